// DiffPoolLayer_40175124087316
// MI455X (gfx1250) — compile-verified
//
#include <hip/hip_runtime.h>
#include <math.h>

typedef __attribute__((ext_vector_type(2))) float v2f;
typedef __attribute__((ext_vector_type(8))) float v8f;

#define NN      32000   // nodes
#define NG      16      // graphs
#define NPG     2000    // nodes per graph
#define KC      1600    // clusters total
#define KPG     100     // clusters per graph
#define DF      128     // feature dim
#define NE      512000  // edges

#define WMMA_F32(a, b, c) \
    __builtin_amdgcn_wmma_f32_16x16x4_f32(false, (a), false, (b), (short)0, (c), false, false)

// ---------------------------------------------------------------------------
// Edge scatter: deg[dst] += 1 ; agg[dst,:] += h[src,:]   (one wave per edge)
// ---------------------------------------------------------------------------
__global__ void edge_agg_kernel(const int* __restrict__ src, const int* __restrict__ dst,
                                const float* __restrict__ h, float* __restrict__ agg,
                                float* __restrict__ deg) {
    int wave = (int)((blockIdx.x * blockDim.x + threadIdx.x) >> 5);
    int lane = threadIdx.x & 31;
    if (wave >= NE) return;
    int s = src[wave], d = dst[wave];
    if (lane == 0) atomicAdd(deg + d, 1.0f);
    const float* hs = h + (size_t)s * DF;
    float* ad = agg + (size_t)d * DF;
#pragma unroll
    for (int i = 0; i < 4; ++i) {
        int f = lane + 32 * i;
        atomicAdd(ad + f, hs[f]);
    }
}

// x = h + agg / max(deg,1)
__global__ void make_x_kernel(const float* __restrict__ h, const float* __restrict__ agg,
                              const float* __restrict__ deg, float* __restrict__ x) {
    int idx = blockIdx.x * blockDim.x + threadIdx.x;
    if (idx >= NN * DF) return;
    int n = idx >> 7;
    float dg = deg[n];
    dg = dg < 1.0f ? 1.0f : dg;
    x[idx] = h[idx] + agg[idx] / dg;
}

// ---------------------------------------------------------------------------
// GEMM + bias + ReLU:  C[M,N] = relu(A[M,K] @ B[K,N] + bias)
// One wave computes a 32x64 tile: 8 WMMA accumulators, two A fragments share
// four B fragments (1.5 loads per WMMA). Software-pipelined k-loop: next
// k-step's fragments are loaded while the current WMMAs execute.
// Requires M%32==0, K%4==0, N%64==0.
// ---------------------------------------------------------------------------
__global__ void gemm_bias_relu(const float* __restrict__ A, const float* __restrict__ B,
                               const float* __restrict__ bias, float* __restrict__ C,
                               int M, int Kd, int Nd) {
    int wave = (int)((blockIdx.x * blockDim.x + threadIdx.x) >> 5);
    int lane = threadIdx.x & 31;
    int tilesN = Nd >> 6;
    int total = (M >> 5) * tilesN;
    if (wave >= total) return;
    int mt = wave / tilesN, nt = wave % tilesN;
    int m0 = mt << 5, n0 = nt << 6;
    int half = lane >> 4, l16 = lane & 15;

    v8f acc[8];
#pragma unroll
    for (int t = 0; t < 8; ++t) acc[t] = (v8f){};

    const float* ar0 = A + (size_t)(m0 + l16) * Kd + 2 * half;
    const float* ar1 = ar0 + (size_t)16 * Kd;
    const float* bp  = B + (size_t)(2 * half) * Nd + n0 + l16;

    v2f a0c, a1c, bc[4];
    a0c.x = ar0[0]; a0c.y = ar0[1];
    a1c.x = ar1[0]; a1c.y = ar1[1];
#pragma unroll
    for (int t = 0; t < 4; ++t) { bc[t].x = bp[t * 16]; bc[t].y = bp[Nd + t * 16]; }

    for (int k0 = 0; k0 < Kd - 4; k0 += 4) {
        v2f a0n, a1n, bn[4];
        const float* an0 = ar0 + k0 + 4;
        const float* an1 = ar1 + k0 + 4;
        const float* bpn = bp + (size_t)(k0 + 4) * Nd;
        a0n.x = an0[0]; a0n.y = an0[1];
        a1n.x = an1[0]; a1n.y = an1[1];
#pragma unroll
        for (int t = 0; t < 4; ++t) { bn[t].x = bpn[t * 16]; bn[t].y = bpn[Nd + t * 16]; }
#pragma unroll
        for (int t = 0; t < 4; ++t) acc[t] = WMMA_F32(a0c, bc[t], acc[t]);
#pragma unroll
        for (int t = 0; t < 4; ++t) acc[4 + t] = WMMA_F32(a1c, bc[t], acc[4 + t]);
        a0c = a0n; a1c = a1n;
#pragma unroll
        for (int t = 0; t < 4; ++t) bc[t] = bn[t];
    }
#pragma unroll
    for (int t = 0; t < 4; ++t) acc[t] = WMMA_F32(a0c, bc[t], acc[t]);
#pragma unroll
    for (int t = 0; t < 4; ++t) acc[4 + t] = WMMA_F32(a1c, bc[t], acc[4 + t]);

    float bb[4];
#pragma unroll
    for (int t = 0; t < 4; ++t) bb[t] = bias[n0 + t * 16 + l16];
#pragma unroll
    for (int r = 0; r < 8; ++r) {
        int row0 = m0 + r + 8 * half;
        float* c0 = C + (size_t)row0 * Nd + n0 + l16;
        float* c1 = c0 + (size_t)16 * Nd;
        float v;
#pragma unroll
        for (int t = 0; t < 4; ++t) {
            v = acc[t][r] + bb[t];     c0[t * 16] = v > 0.0f ? v : 0.0f;
            v = acc[4 + t][r] + bb[t]; c1[t * 16] = v > 0.0f ? v : 0.0f;
        }
    }
}

// ---------------------------------------------------------------------------
// Fused assign layer 2 (block-compacted columns) + masked softmax.
// Wave computes 16 node-rows x 112 cols (100 valid) of relu(a1 @ W2p + b2p),
// then the exact reference masked-softmax:
//   m = max(row,0); S = sum_in exp(v-m); Z = S + 1500*exp(-m)
//   out = (exp(v-m)/Z) / (S/Z + 1e-13)
// Result stored compacted: assignc[N,100]. K-loop software-pipelined.
// ---------------------------------------------------------------------------
__global__ void assign_l2_softmax(const float* __restrict__ A /*[NN,KC]*/,
                                  const float* __restrict__ W /*[KC,KC]*/,
                                  const float* __restrict__ bias /*[KC]*/,
                                  float* __restrict__ assignc /*[NN,KPG]*/) {
    int wave = (int)((blockIdx.x * blockDim.x + threadIdx.x) >> 5);
    int lane = threadIdx.x & 31;
    if (wave >= NN / 16) return;
    int m0 = wave << 4;
    int gb = m0 / NPG;
    int colbase = gb * KPG;
    int half = lane >> 4, l16 = lane & 15;

    int  cloc[7];
    bool cvalid[7];
    int  bcol[7];
#pragma unroll
    for (int t = 0; t < 7; ++t) {
        cloc[t] = t * 16 + l16;
        cvalid[t] = cloc[t] < KPG;
        bcol[t] = colbase + (cvalid[t] ? cloc[t] : 0);
    }

    v8f acc[7];
#pragma unroll
    for (int t = 0; t < 7; ++t) acc[t] = (v8f){};

    const float* arow  = A + (size_t)(m0 + l16) * KC + 2 * half;
    const float* bbase = W + (size_t)(2 * half) * KC;

    v2f ac, bc[7];
    ac.x = arow[0]; ac.y = arow[1];
#pragma unroll
    for (int t = 0; t < 7; ++t) { bc[t].x = bbase[bcol[t]]; bc[t].y = bbase[KC + bcol[t]]; }

    for (int k0 = 0; k0 < KC - 4; k0 += 4) {
        v2f an, bn[7];
        const float* ap = arow + k0 + 4;
        const float* bp = bbase + (size_t)(k0 + 4) * KC;
        an.x = ap[0]; an.y = ap[1];
#pragma unroll
        for (int t = 0; t < 7; ++t) { bn[t].x = bp[bcol[t]]; bn[t].y = bp[KC + bcol[t]]; }
#pragma unroll
        for (int t = 0; t < 7; ++t) acc[t] = WMMA_F32(ac, bc[t], acc[t]);
        ac = an;
#pragma unroll
        for (int t = 0; t < 7; ++t) bc[t] = bn[t];
    }
#pragma unroll
    for (int t = 0; t < 7; ++t) acc[t] = WMMA_F32(ac, bc[t], acc[t]);

#pragma unroll
    for (int r = 0; r < 8; ++r) {
        int row = m0 + r + 8 * half;
        float vals[7];
        float mymax = 0.0f;  // reference max over masked row includes the zeros
#pragma unroll
        for (int t = 0; t < 7; ++t) {
            float v = acc[t][r] + bias[bcol[t]];
            v = v > 0.0f ? v : 0.0f;          // relu
            vals[t] = cvalid[t] ? v : 0.0f;
            mymax = fmaxf(mymax, vals[t]);
        }
#pragma unroll
        for (int off = 1; off < 16; off <<= 1)
            mymax = fmaxf(mymax, __shfl_xor(mymax, off, 32));
        float es[7];
        float s = 0.0f;
#pragma unroll
        for (int t = 0; t < 7; ++t) {
            es[t] = expf(vals[t] - mymax);
            if (cvalid[t]) s += es[t];
        }
#pragma unroll
        for (int off = 1; off < 16; off <<= 1)
            s += __shfl_xor(s, off, 32);
        float Z = s + (float)(KC - KPG) * expf(-mymax);
        float denom = s / Z + 1e-13f;
        float scale = 1.0f / (Z * denom);
#pragma unroll
        for (int t = 0; t < 7; ++t)
            if (cvalid[t])
                assignc[(size_t)row * KPG + cloc[t]] = es[t] * scale;
    }
}

// ---------------------------------------------------------------------------
// h_pool block: out[gb*100+j, d] = sum_n assignc[n,j] * feat[n,d]  (per graph)
// Wave: 16(clusters) x 64(feat) tile, A loaded transposed from assignc.
// Waves: NG * 7 * 2 = 224. Pipelined k-loop.
// ---------------------------------------------------------------------------
__global__ void hpool_kernel(const float* __restrict__ assignc, const float* __restrict__ feat,
                             float* __restrict__ out /* d_out + KC*KC */) {
    int wave = (int)((blockIdx.x * blockDim.x + threadIdx.x) >> 5);
    int lane = threadIdx.x & 31;
    if (wave >= NG * 14) return;
    int gb = wave / 14;
    int rem = wave % 14;
    int mt = rem >> 1, nt = rem & 1;
    int m0 = mt << 4, n0 = nt << 6;
    int half = lane >> 4, l16 = lane & 15;

    int j = m0 + l16;
    int jc = j < KPG ? j : (KPG - 1);
    size_t nb = (size_t)gb * NPG;

    const float* ap = assignc + (nb + 2 * half) * KPG + jc;   // step: 4*KPG
    const float* bp = feat + (nb + 2 * half) * DF + n0 + l16; // step: 4*DF

    v8f acc[4];
#pragma unroll
    for (int t = 0; t < 4; ++t) acc[t] = (v8f){};

    v2f ac, bc[4];
    ac.x = ap[0]; ac.y = ap[KPG];
#pragma unroll
    for (int t = 0; t < 4; ++t) { bc[t].x = bp[t * 16]; bc[t].y = bp[DF + t * 16]; }

    for (int k0 = 0; k0 < NPG - 4; k0 += 4) {
        v2f an, bn[4];
        const float* apn = ap + (size_t)(k0 + 4) * KPG;
        const float* bpn = bp + (size_t)(k0 + 4) * DF;
        an.x = apn[0]; an.y = apn[KPG];
#pragma unroll
        for (int t = 0; t < 4; ++t) { bn[t].x = bpn[t * 16]; bn[t].y = bpn[DF + t * 16]; }
#pragma unroll
        for (int t = 0; t < 4; ++t) acc[t] = WMMA_F32(ac, bc[t], acc[t]);
        ac = an;
#pragma unroll
        for (int t = 0; t < 4; ++t) bc[t] = bn[t];
    }
#pragma unroll
    for (int t = 0; t < 4; ++t) acc[t] = WMMA_F32(ac, bc[t], acc[t]);

#pragma unroll
    for (int r = 0; r < 8; ++r) {
        int jj = m0 + r + 8 * half;
        if (jj < KPG) {
            float* o = out + (size_t)(gb * KPG + jj) * DF + n0 + l16;
#pragma unroll
            for (int t = 0; t < 4; ++t) o[t * 16] = acc[t][r];
        }
    }
}

// tmpc[dst,:] += assignc[src,:]  (one wave per edge, 100 cols)
__global__ void tmp_scatter_kernel(const int* __restrict__ src, const int* __restrict__ dst,
                                   const float* __restrict__ assignc, float* __restrict__ tmpc) {
    int wave = (int)((blockIdx.x * blockDim.x + threadIdx.x) >> 5);
    int lane = threadIdx.x & 31;
    if (wave >= NE) return;
    int s = src[wave], d = dst[wave];
    const float* as = assignc + (size_t)s * KPG;
    float* td = tmpc + (size_t)d * KPG;
#pragma unroll
    for (int i = 0; i < 4; ++i) {
        int jj = lane + 32 * i;
        if (jj < KPG) atomicAdd(td + jj, as[jj]);
    }
}

// ---------------------------------------------------------------------------
// adj block: out[gb*100+i, gb*100+j] = sum_n assignc[n,i] * tmpc[n,j]
// Wave: 16(i) x 112(j, 100 valid) tile. Waves: NG*7 = 112. Pipelined k-loop.
// ---------------------------------------------------------------------------
__global__ void adj_kernel(const float* __restrict__ assignc, const float* __restrict__ tmpc,
                           float* __restrict__ out /* [KC,KC] */) {
    int wave = (int)((blockIdx.x * blockDim.x + threadIdx.x) >> 5);
    int lane = threadIdx.x & 31;
    if (wave >= NG * 7) return;
    int gb = wave / 7, mt = wave % 7;
    int m0 = mt << 4;
    int half = lane >> 4, l16 = lane & 15;

    int i = m0 + l16;
    int ic = i < KPG ? i : (KPG - 1);
    int  cloc[7];
    bool cvalid[7];
    int  cc[7];
#pragma unroll
    for (int t = 0; t < 7; ++t) {
        cloc[t] = t * 16 + l16;
        cvalid[t] = cloc[t] < KPG;
        cc[t] = cvalid[t] ? cloc[t] : 0;
    }
    size_t nb = (size_t)gb * NPG;

    const float* ap = assignc + (nb + 2 * half) * KPG + ic;  // step 4*KPG
    const float* bp = tmpc + (nb + 2 * half) * KPG;          // step 4*KPG

    v8f acc[7];
#pragma unroll
    for (int t = 0; t < 7; ++t) acc[t] = (v8f){};

    v2f ac, bc[7];
    ac.x = ap[0]; ac.y = ap[KPG];
#pragma unroll
    for (int t = 0; t < 7; ++t) { bc[t].x = bp[cc[t]]; bc[t].y = bp[KPG + cc[t]]; }

    for (int k0 = 0; k0 < NPG - 4; k0 += 4) {
        v2f an, bn[7];
        const float* apn = ap + (size_t)(k0 + 4) * KPG;
        const float* bpn = bp + (size_t)(k0 + 4) * KPG;
        an.x = apn[0]; an.y = apn[KPG];
#pragma unroll
        for (int t = 0; t < 7; ++t) { bn[t].x = bpn[cc[t]]; bn[t].y = bpn[KPG + cc[t]]; }
#pragma unroll
        for (int t = 0; t < 7; ++t) acc[t] = WMMA_F32(ac, bc[t], acc[t]);
        ac = an;
#pragma unroll
        for (int t = 0; t < 7; ++t) bc[t] = bn[t];
    }
#pragma unroll
    for (int t = 0; t < 7; ++t) acc[t] = WMMA_F32(ac, bc[t], acc[t]);

#pragma unroll
    for (int r = 0; r < 8; ++r) {
        int ii = m0 + r + 8 * half;
        if (ii < KPG) {
            float* orow = out + (size_t)(gb * KPG + ii) * KC + gb * KPG;
#pragma unroll
            for (int t = 0; t < 7; ++t)
                if (cvalid[t]) orow[cloc[t]] = acc[t][r];
        }
    }
}

// ---------------------------------------------------------------------------
extern "C" void kernel_launch(void* const* d_in, const int* in_sizes, int n_in,
                              void* d_out, int out_size, void* d_ws, size_t ws_size,
                              hipStream_t stream) {
    const float* h   = (const float*)d_in[0];
    const int*   src = (const int*)d_in[1];
    const int*   dst = (const int*)d_in[2];
    const float* W1f = (const float*)d_in[3];
    const float* b1f = (const float*)d_in[4];
    const float* W2f = (const float*)d_in[5];
    const float* b2f = (const float*)d_in[6];
    const float* W1p = (const float*)d_in[7];
    const float* b1p = (const float*)d_in[8];
    const float* W2p = (const float*)d_in[9];
    const float* b2p = (const float*)d_in[10];
    float* out = (float*)d_out;

    // Workspace layout (floats). Atomically-accumulated buffers first so one
    // memset covers them. Total ~296 MB.
    float* ws = (float*)d_ws;
    size_t off = 0;
    float* agg     = ws + off; off += (size_t)NN * DF;   // 4.096M
    float* deg     = ws + off; off += (size_t)NN;        // 32K
    float* tmpc    = ws + off; off += (size_t)NN * KPG;  // 3.2M
    size_t zeroFloats = off;
    float* x       = ws + off; off += (size_t)NN * DF;
    float* y1      = ws + off; off += (size_t)NN * DF;
    float* feat    = ws + off; off += (size_t)NN * DF;
    float* a1      = ws + off; off += (size_t)NN * KC;   // 51.2M
    float* assignc = ws + off; off += (size_t)NN * KPG;

    hipMemsetAsync(ws, 0, zeroFloats * sizeof(float), stream);
    hipMemsetAsync(out, 0, (size_t)KC * KC * sizeof(float), stream); // adj block-diag fill

    // 1. edge aggregation (deg + agg)
    edge_agg_kernel<<<NE / 8, 256, 0, stream>>>(src, dst, h, agg, deg);

    // 2. x = h + agg / max(deg,1)
    make_x_kernel<<<(NN * DF) / 256, 256, 0, stream>>>(h, agg, deg, x);

    // 3. feat MLP (two 128x128 layers), 32x64 tiles
    {
        int waves = (NN / 32) * (DF / 64);                  // 2000
        gemm_bias_relu<<<(waves + 7) / 8, 256, 0, stream>>>(x, W1f, b1f, y1, NN, DF, DF);
        gemm_bias_relu<<<(waves + 7) / 8, 256, 0, stream>>>(y1, W2f, b2f, feat, NN, DF, DF);
    }

    // 4. assign layer 1: a1 = relu(x @ W1p + b1p)   [NN, 1600]
    {
        int waves = (NN / 32) * (KC / 64);                  // 25000
        gemm_bias_relu<<<(waves + 7) / 8, 256, 0, stream>>>(x, W1p, b1p, a1, NN, DF, KC);
    }

    // 5. fused compacted layer 2 + masked softmax -> assignc [NN,100]
    {
        int waves = NN / 16;                                // 2000
        assign_l2_softmax<<<(waves + 7) / 8, 256, 0, stream>>>(a1, W2p, b2p, assignc);
    }

    // 6. h_pool -> d_out[KC*KC : KC*KC + KC*DF]
    {
        int waves = NG * 14;                                // 224
        hpool_kernel<<<(waves + 7) / 8, 256, 0, stream>>>(assignc, feat, out + (size_t)KC * KC);
    }

    // 7. tmp = scatter-add of assignc over edges
    tmp_scatter_kernel<<<NE / 8, 256, 0, stream>>>(src, dst, assignc, tmpc);

    // 8. adj_new diagonal blocks -> d_out[0 : KC*KC]
    {
        int waves = NG * 7;                                 // 112
        adj_kernel<<<(waves + 7) / 8, 256, 0, stream>>>(assignc, tmpc, out);
    }
}